// HUG_MHE_35811437314877
// MI455X (gfx1250) — compile-verified
//
#include <hip/hip_runtime.h>

#define ALPHA 1.0f
#define BETA  1.0f

constexpr int N  = 131072;
constexpr int D  = 512;
constexpr int C  = 1000;
constexpr int CP = 1024;       // class count padded to tile multiple
constexpr int T  = CP / 16;    // 64 tiles per side

typedef float v2f __attribute__((ext_vector_type(2)));
typedef float v4f __attribute__((ext_vector_type(4)));
typedef float v8f __attribute__((ext_vector_type(8)));

// workspace layout in floats
constexpr size_t W_OFF   = 0;                      // CP*D normalized (zero-padded) weights
constexpr size_t SQ_OFF  = W_OFF + (size_t)CP * D; // CP   row squared norms of w
constexpr size_t SUM_OFF = SQ_OFF + CP;            // CP   per-class dist sums
constexpr size_t CNT_OFF = SUM_OFF + CP;           // CP   per-class counts
constexpr size_t ACC_OFF = CNT_OFF + CP;           // 2    [sum(1/d2), mask count]

__device__ __forceinline__ float wave_sum(float v) {
  #pragma unroll
  for (int off = 16; off >= 1; off >>= 1)
    v += __shfl_xor(v, off, 32);
  return v;
}

// --- zero the accumulators (ws is poisoned by harness, never re-zeroed) ---
__global__ void k_zero(float* __restrict__ ws) {
  const int i = threadIdx.x;          // blockDim == 1024 == CP
  ws[SUM_OFF + i] = 0.0f;
  ws[CNT_OFF + i] = 0.0f;
  if (i < 2) ws[ACC_OFF + i] = 0.0f;
}

// --- normalize classifier rows into padded w; one wave per row ---
__global__ void k_norm(const float* __restrict__ cls, float* __restrict__ ws) {
  const int wave = threadIdx.x >> 5;
  const int lane = threadIdx.x & 31;
  const int row  = blockIdx.x * 8 + wave;            // grid = CP/8
  float* wout = ws + W_OFF + (size_t)row * D + lane * 16;
  if (row >= C) {                                    // zero padding rows
    const v4f z = {0.f, 0.f, 0.f, 0.f};
    #pragma unroll
    for (int q = 0; q < 4; ++q) *(v4f*)(wout + q * 4) = z;
    if (lane == 0) ws[SQ_OFF + row] = 0.0f;
    return;
  }
  const float* src = cls + (size_t)row * D + lane * 16;
  v4f x[4];
  float ss = 0.0f;
  #pragma unroll
  for (int q = 0; q < 4; ++q) {
    x[q] = *(const v4f*)(src + q * 4);
    ss += x[q].x * x[q].x + x[q].y * x[q].y + x[q].z * x[q].z + x[q].w * x[q].w;
  }
  ss = wave_sum(ss);
  const float inv = 1.0f / sqrtf(ss);
  #pragma unroll
  for (int q = 0; q < 4; ++q) {
    v4f y = {x[q].x * inv, x[q].y * inv, x[q].z * inv, x[q].w * inv};
    *(v4f*)(wout + q * 4) = y;
  }
  if (lane == 0) ws[SQ_OFF + row] = ss * inv * inv;  // == sum(w*w) per reference
}

// --- Gram matrix via V_WMMA_F32_16X16X4_F32; one wave per 16x16 upper tile ---
__global__ void k_gram(const float* __restrict__ w, const float* __restrict__ sq,
                       float* __restrict__ acc) {
  const int ti = blockIdx.x / T;
  const int tj = blockIdx.x % T;
  if (tj < ti) return;                               // strict upper-triangle tiles (+diag)
  const int lane = threadIdx.x;                      // blockDim == 32, EXEC all-ones
  const int m  = lane & 15;                          // A row / B col within tile
  const int kb = (lane >> 4) * 2;                    // K-pair select per ISA layout
  const float* pa = w + (size_t)(ti * 16 + m) * D + kb;
  const float* pb = w + (size_t)(tj * 16 + m) * D + kb;
  v8f c = {};
  #pragma unroll 4
  for (int k0 = 0; k0 < D; k0 += 4) {
    v2f a = *(const v2f*)(pa + k0);
    v2f b = *(const v2f*)(pb + k0);
    c = __builtin_amdgcn_wmma_f32_16x16x4_f32(false, a, false, b, (short)0, c,
                                              false, false);
  }
  // C/D layout: lane%16 = column, VGPR v = row v (+8 for upper lane half)
  const int col = lane & 15;
  const int rb  = (lane >> 4) * 8;
  float sumInv = 0.0f, cnt = 0.0f;
  #pragma unroll
  for (int v = 0; v < 8; ++v) {
    const int i = ti * 16 + rb + v;
    const int j = tj * 16 + col;
    if (i < C && j < C && j > i) {
      float d2 = fmaxf(sq[i] + sq[j] - 2.0f * c[v], 0.0f);
      if (d2 > 0.0f) { sumInv += 1.0f / d2; cnt += 1.0f; }
    }
  }
  atomicAdd(&acc[0], sumInv);
  atomicAdd(&acc[1], cnt);
}

// --- fused: feat passthrough copy + row norm + dot(feat_n, w[target]) + class sums ---
__global__ void k_sample(const float* __restrict__ feat, const int* __restrict__ target,
                         const float* __restrict__ w, const float* __restrict__ sq,
                         float* __restrict__ outF, float* __restrict__ sums,
                         float* __restrict__ cnts) {
  const int wave = threadIdx.x >> 5;
  const int lane = threadIdx.x & 31;
  const int row  = blockIdx.x * 8 + wave;            // grid = N/8
  const int t    = target[row];
  const float* f    = feat + (size_t)row * D + lane * 16;
  const float* wrow = w    + (size_t)t   * D + lane * 16;
  float ss = 0.0f, dt = 0.0f;
  v4f x[4];
  #pragma unroll
  for (int q = 0; q < 4; ++q) {
    x[q] = __builtin_nontemporal_load((const v4f*)(f + q * 4));
    const v4f wv = *(const v4f*)(wrow + q * 4);      // w is hot in L2 (2 MB)
    ss += x[q].x * x[q].x + x[q].y * x[q].y + x[q].z * x[q].z + x[q].w * x[q].w;
    dt += x[q].x * wv.x   + x[q].y * wv.y   + x[q].z * wv.z   + x[q].w * wv.w;
  }
  float* o = outF + (size_t)row * D + lane * 16;     // only 8B-aligned (d_out+2)
  #pragma unroll
  for (int q = 0; q < 4; ++q) {
    v2f lo = {x[q].x, x[q].y}, hi = {x[q].z, x[q].w};
    __builtin_nontemporal_store(lo, (v2f*)(o + q * 4));
    __builtin_nontemporal_store(hi, (v2f*)(o + q * 4 + 2));
  }
  ss = wave_sum(ss);
  dt = wave_sum(dt);
  if (lane == 0) {
    const float inv  = 1.0f / sqrtf(ss);
    const float d2   = ss * inv * inv + sq[t] - 2.0f * dt * inv;
    const float dist = sqrtf(fmaxf(d2, 0.0f));
    atomicAdd(&sums[t], dist);
    atomicAdd(&cnts[t], 1.0f);
  }
}

// --- final scalar reductions; single wave ---
__global__ void k_final(const float* __restrict__ sums, const float* __restrict__ cnts,
                        const float* __restrict__ acc, float* __restrict__ out) {
  const int lane = threadIdx.x;                      // blockDim == 32
  float s = 0.0f, p = 0.0f;
  for (int c = lane; c < C; c += 32) {
    const float n = cnts[c];
    if (n > 0.0f) { s += sums[c] / n; p += 1.0f; }
  }
  s = wave_sum(s);
  p = wave_sum(p);
  if (lane == 0) {
    out[0] = (s / p) * BETA;            // sample_wise_all * BETA
    out[1] = (acc[0] / acc[1]) * ALPHA; // weight_wise * ALPHA
  }
}

extern "C" void kernel_launch(void* const* d_in, const int* in_sizes, int n_in,
                              void* d_out, int out_size, void* d_ws, size_t ws_size,
                              hipStream_t stream) {
  (void)in_sizes; (void)n_in; (void)out_size; (void)ws_size;
  const float* feat = (const float*)d_in[0];
  const float* cls  = (const float*)d_in[1];
  const int*   tgt  = (const int*)d_in[2];
  float* out = (float*)d_out;
  float* ws  = (float*)d_ws;   // needs ~2.11 MB

  k_zero  <<<1,        1024, 0, stream>>>(ws);
  k_norm  <<<CP / 8,   256,  0, stream>>>(cls, ws);
  k_gram  <<<T * T,    32,   0, stream>>>(ws + W_OFF, ws + SQ_OFF, ws + ACC_OFF);
  k_sample<<<N / 8,    256,  0, stream>>>(feat, tgt, ws + W_OFF, ws + SQ_OFF,
                                          out + 2, ws + SUM_OFF, ws + CNT_OFF);
  k_final <<<1,        32,   0, stream>>>(ws + SUM_OFF, ws + CNT_OFF, ws + ACC_OFF, out);
}